// Attention_3487513445039
// MI455X (gfx1250) — compile-verified
//
#include <hip/hip_runtime.h>
#include <hip/hip_bf16.h>

// ---------------------------------------------------------------------------
// Performer (FAVOR+) causal linear attention, ReLU random features, fp32.
// B=2, L=4096, H=8, D=64, M=256.  3-phase chunked-scan formulation so the
// whole chip is busy despite the sequential causal recurrence.
// All matmuls via V_WMMA_F32_16X16X4_F32 (fp32 in/out — kernel is HBM-bound,
// so fp32 WMMA is free).  Next-chunk global_prefetch_b8 hides HBM latency
// across the barriered pipeline; output uses non-temporal stores.
// ---------------------------------------------------------------------------

#define Bd    2
#define Hd    8
#define Ld    4096
#define Dd    64
#define Md    256
#define SEG   256                 // rows per segment (phase-1/3 workgroup)
#define NSEG  (Ld / SEG)          // 16
#define CK    32                  // chunk rows inside a segment
#define NCH   (SEG / CK)          // 8
#define PSTR  65                  // LDS stride for [*][64] tiles (pad -> no bank conflicts)
#define FSTR  257                 // LDS stride for [*][256] feature tiles
#define ASTR  33                  // LDS stride for [32][32] attention tile
#define STATE (Md * Dd + Md)      // per-segment state floats (S + s)
#define RATIO 0.0625f             // 1/sqrt(M)
#define EPS   0.001f              // NUM_STAB

typedef float v8f __attribute__((ext_vector_type(8)));
typedef float v2f __attribute__((ext_vector_type(2)));

static __device__ __forceinline__ v8f wmma4(v2f a, v2f b, v8f c) {
  // D[16x16] = A[16x4] * B[4x16] + C   (fp32)
  return __builtin_amdgcn_wmma_f32_16x16x4_f32(
      /*neg_a=*/false, a, /*neg_b=*/false, b,
      /*c_mod=*/(short)0, c, /*reuse_a=*/false, /*reuse_b=*/false);
}

// Prefetch one CK-row chunk of up to three [B,L,H,D] tensors for (b,h).
// 32 rows x 256B each; 2 prefetches per row (128B apart).
static __device__ __forceinline__ void prefetch_chunk(
    const float* t0, const float* t1, const float* t2,
    int ntens, int b, int h, int lbase, int tid) {
  const int per = CK * 2;                      // 64 prefetches per tensor
  if (tid < ntens * per) {
    int which = tid / per;
    int r     = (tid % per) >> 1;
    int halfo = (tid & 1) * 32;                // float offset (128B)
    const float* src = (which == 0) ? t0 : (which == 1) ? t1 : t2;
    const float* addr = src + (((size_t)b * Ld + lbase + r) * Hd + h) * Dd + halfo;
    __builtin_prefetch(addr, 0, 3);            // -> global_prefetch_b8
  }
}

// ---------------------------------------------------------------------------
// Phase 1: per-segment sums  S_seg = Kf^T V ,  s_seg = colsum(Kf)
// grid = B*H*NSEG blocks, 256 threads (8 wave32)
// ---------------------------------------------------------------------------
__global__ __launch_bounds__(256) void perf_attn_segsum(
    const float* __restrict__ kk, const float* __restrict__ vv,
    const float* __restrict__ proj, float* __restrict__ ws)
{
  __shared__ float Plds[Md * PSTR];   // projection [M][D], padded
  __shared__ float Slds[Md * PSTR];   // running K^T V
  __shared__ float sLds[Md];
  __shared__ float raw [CK * PSTR];   // K chunk
  __shared__ float Vlds[CK * PSTR];   // V chunk
  __shared__ float Kf  [CK * FSTR];   // K features

  const int tid  = threadIdx.x;
  const int lane = tid & 31, wave = tid >> 5;
  const int lm   = lane & 15, hi   = lane >> 4;

  const int bh  = blockIdx.x / NSEG;
  const int seg = blockIdx.x % NSEG;
  const int b = bh / Hd, h = bh % Hd;

  for (int i = tid; i < Md * Dd; i += 256) {
    Plds[(i >> 6) * PSTR + (i & 63)] = proj[i];
    Slds[(i >> 6) * PSTR + (i & 63)] = 0.f;
  }
  for (int i = tid; i < Md; i += 256) sLds[i] = 0.f;
  __syncthreads();

  for (int ci = 0; ci < NCH; ++ci) {
    const int lbase = seg * SEG + ci * CK;
    for (int i = tid; i < CK * Dd; i += 256) {
      int r = i >> 6, d = i & 63;
      size_t g = (((size_t)b * Ld + lbase + r) * Hd + h) * Dd + d;
      raw [r * PSTR + d] = kk[g];
      Vlds[r * PSTR + d] = vv[g];
    }
    // overlap: prefetch next chunk's K/V while this chunk computes
    if (ci + 1 < NCH)
      prefetch_chunk(kk, vv, vv, 2, b, h, lbase + CK, tid);
    __syncthreads();

    // Kf[32][256] = relu(RATIO * raw @ P^T) + EPS
    for (int t = wave; t < 2 * (Md / 16); t += 8) {
      int it = t & 1, mt = t >> 1;
      int r0 = it * 16, c0 = mt * 16;
      v8f acc = {};
      for (int k0 = 0; k0 < Dd; k0 += 4) {
        v2f a, bb;
        a.x  = raw [(r0 + lm) * PSTR + k0 + 2 * hi];
        a.y  = raw [(r0 + lm) * PSTR + k0 + 2 * hi + 1];
        bb.x = Plds[(c0 + lm) * PSTR + k0 + 2 * hi];
        bb.y = Plds[(c0 + lm) * PSTR + k0 + 2 * hi + 1];
        acc = wmma4(a, bb, acc);
      }
      for (int e = 0; e < 8; ++e)
        Kf[(r0 + e + 8 * hi) * FSTR + c0 + lm] =
            fmaxf(RATIO * acc[e], 0.f) + EPS;
    }
    __syncthreads();

    // s += colsum(Kf)
    {
      float cs = 0.f;
      for (int r = 0; r < CK; ++r) cs += Kf[r * FSTR + tid];
      sLds[tid] += cs;
    }
    // S += Kf^T @ V   (64 tiles of 16x16, 8 per wave)
    for (int t = wave * 8; t < wave * 8 + 8; ++t) {
      int mt = t >> 2, dt = t & 3;
      int r0 = mt * 16, c0 = dt * 16;
      v8f acc;
      for (int e = 0; e < 8; ++e)
        acc[e] = Slds[(r0 + e + 8 * hi) * PSTR + c0 + lm];
      for (int j0 = 0; j0 < CK; j0 += 4) {
        v2f a, bb;
        a.x  = Kf  [(j0 + 2 * hi)     * FSTR + r0 + lm];
        a.y  = Kf  [(j0 + 2 * hi + 1) * FSTR + r0 + lm];
        bb.x = Vlds[(j0 + 2 * hi)     * PSTR + c0 + lm];
        bb.y = Vlds[(j0 + 2 * hi + 1) * PSTR + c0 + lm];
        acc = wmma4(a, bb, acc);
      }
      for (int e = 0; e < 8; ++e)
        Slds[(r0 + e + 8 * hi) * PSTR + c0 + lm] = acc[e];
    }
    __syncthreads();
  }

  float* dst = ws + (size_t)(bh * NSEG + seg) * STATE;
  for (int i = tid; i < Md * Dd; i += 256)
    dst[i] = Slds[(i >> 6) * PSTR + (i & 63)];
  for (int i = tid; i < Md; i += 256)
    dst[Md * Dd + i] = sLds[i];
}

// ---------------------------------------------------------------------------
// Phase 2: exclusive prefix over segments (per b,h).  grid = B*H blocks.
// ---------------------------------------------------------------------------
__global__ __launch_bounds__(256) void perf_attn_prefix(float* __restrict__ ws)
{
  const int bh = blockIdx.x;
  float* base = ws + (size_t)bh * NSEG * STATE;
  for (int e = threadIdx.x; e < STATE; e += 256) {
    float run = 0.f;
    for (int sgi = 0; sgi < NSEG; ++sgi) {
      float* p = base + (size_t)sgi * STATE + e;
      float t = *p;
      *p = run;
      run += t;
    }
  }
}

// ---------------------------------------------------------------------------
// Phase 3: per-segment causal output using exclusive-prefix state.
// grid = B*H*NSEG blocks, 256 threads (8 wave32)
// ---------------------------------------------------------------------------
__global__ __launch_bounds__(256) void perf_attn_main(
    const float* __restrict__ qq, const float* __restrict__ kk,
    const float* __restrict__ vv, const float* __restrict__ proj,
    const float* __restrict__ ws, float* __restrict__ out)
{
  __shared__ float Plds[Md * PSTR];
  __shared__ float Slds[Md * PSTR];
  __shared__ float sLds[Md];
  __shared__ float raw [CK * PSTR];   // Q chunk, then reused for K chunk
  __shared__ float Vlds[CK * PSTR];
  __shared__ float Qf  [CK * FSTR];
  __shared__ float Kf  [CK * FSTR];
  __shared__ float Alds[CK * ASTR];   // masked Qf @ Kf^T
  __shared__ float denLds[CK];

  const int tid  = threadIdx.x;
  const int lane = tid & 31, wave = tid >> 5;
  const int lm   = lane & 15, hi   = lane >> 4;

  const int bh  = blockIdx.x / NSEG;
  const int seg = blockIdx.x % NSEG;
  const int b = bh / Hd, h = bh % Hd;

  for (int i = tid; i < Md * Dd; i += 256)
    Plds[(i >> 6) * PSTR + (i & 63)] = proj[i];
  const float* st = ws + (size_t)(bh * NSEG + seg) * STATE;
  for (int i = tid; i < Md * Dd; i += 256)
    Slds[(i >> 6) * PSTR + (i & 63)] = st[i];
  for (int i = tid; i < Md; i += 256) sLds[i] = st[Md * Dd + i];
  __syncthreads();

  for (int ci = 0; ci < NCH; ++ci) {
    const int lbase = seg * SEG + ci * CK;

    // ---- stage A: load Q chunk; prefetch next chunk's Q/K/V
    for (int i = tid; i < CK * Dd; i += 256) {
      int r = i >> 6, d = i & 63;
      raw[r * PSTR + d] = qq[(((size_t)b * Ld + lbase + r) * Hd + h) * Dd + d];
    }
    if (ci + 1 < NCH)
      prefetch_chunk(qq, kk, vv, 3, b, h, lbase + CK, tid);
    __syncthreads();

    // ---- stage B: Qf features
    for (int t = wave; t < 2 * (Md / 16); t += 8) {
      int it = t & 1, mt = t >> 1;
      int r0 = it * 16, c0 = mt * 16;
      v8f acc = {};
      for (int k0 = 0; k0 < Dd; k0 += 4) {
        v2f a, bb;
        a.x  = raw [(r0 + lm) * PSTR + k0 + 2 * hi];
        a.y  = raw [(r0 + lm) * PSTR + k0 + 2 * hi + 1];
        bb.x = Plds[(c0 + lm) * PSTR + k0 + 2 * hi];
        bb.y = Plds[(c0 + lm) * PSTR + k0 + 2 * hi + 1];
        acc = wmma4(a, bb, acc);
      }
      for (int e = 0; e < 8; ++e)
        Qf[(r0 + e + 8 * hi) * FSTR + c0 + lm] =
            fmaxf(RATIO * acc[e], 0.f) + EPS;
    }
    __syncthreads();

    // ---- stage C: load K chunk (reuse raw) and V chunk
    for (int i = tid; i < CK * Dd; i += 256) {
      int r = i >> 6, d = i & 63;
      size_t g = (((size_t)b * Ld + lbase + r) * Hd + h) * Dd + d;
      raw [r * PSTR + d] = kk[g];
      Vlds[r * PSTR + d] = vv[g];
    }
    __syncthreads();

    // ---- stage D: Kf features
    for (int t = wave; t < 2 * (Md / 16); t += 8) {
      int it = t & 1, mt = t >> 1;
      int r0 = it * 16, c0 = mt * 16;
      v8f acc = {};
      for (int k0 = 0; k0 < Dd; k0 += 4) {
        v2f a, bb;
        a.x  = raw [(r0 + lm) * PSTR + k0 + 2 * hi];
        a.y  = raw [(r0 + lm) * PSTR + k0 + 2 * hi + 1];
        bb.x = Plds[(c0 + lm) * PSTR + k0 + 2 * hi];
        bb.y = Plds[(c0 + lm) * PSTR + k0 + 2 * hi + 1];
        acc = wmma4(a, bb, acc);
      }
      for (int e = 0; e < 8; ++e)
        Kf[(r0 + e + 8 * hi) * FSTR + c0 + lm] =
            fmaxf(RATIO * acc[e], 0.f) + EPS;
    }
    __syncthreads();

    // ---- stage E: A = causal_mask(Qf @ Kf^T)  (3 live tiles; (0,1) zeroed)
    if (wave < 3) {
      const int itA[3] = {0, 1, 1};
      const int jtA[3] = {0, 0, 1};
      int it = itA[wave], jt = jtA[wave];
      v8f acc = {};
      for (int k0 = 0; k0 < Md; k0 += 4) {
        v2f a, bb;
        a.x  = Qf[(it * 16 + lm) * FSTR + k0 + 2 * hi];
        a.y  = Qf[(it * 16 + lm) * FSTR + k0 + 2 * hi + 1];
        bb.x = Kf[(jt * 16 + lm) * FSTR + k0 + 2 * hi];
        bb.y = Kf[(jt * 16 + lm) * FSTR + k0 + 2 * hi + 1];
        acc = wmma4(a, bb, acc);
      }
      for (int e = 0; e < 8; ++e) {
        int gi = it * 16 + e + 8 * hi, gj = jt * 16 + lm;
        Alds[gi * ASTR + gj] = (gj <= gi) ? acc[e] : 0.f;  // inclusive causal
      }
    } else if (wave == 3) {
      for (int i2 = lane; i2 < 256; i2 += 32)
        Alds[(i2 >> 4) * ASTR + 16 + (i2 & 15)] = 0.f;
    }
    __syncthreads();

    // ---- stage F: den (wave 0) + out accumulators (tile per wave)
    if (wave == 0) {
      float dsum = 0.f;
      for (int m = 0; m < Md; ++m) dsum += Qf[lane * FSTR + m] * sLds[m];
      for (int j = 0; j < CK; ++j) dsum += Alds[lane * ASTR + j];
      denLds[lane] = dsum;
    }
    v8f oacc = {};
    {
      int it = wave >> 2, dt = wave & 3;
      int r0 = it * 16, c0 = dt * 16;
      for (int k0 = 0; k0 < Md; k0 += 4) {        // inter: Qf @ S_prev
        v2f a, bb;
        a.x  = Qf  [(r0 + lm) * FSTR + k0 + 2 * hi];
        a.y  = Qf  [(r0 + lm) * FSTR + k0 + 2 * hi + 1];
        bb.x = Slds[(k0 + 2 * hi)     * PSTR + c0 + lm];
        bb.y = Slds[(k0 + 2 * hi + 1) * PSTR + c0 + lm];
        oacc = wmma4(a, bb, oacc);
      }
      for (int j0 = 0; j0 < CK; j0 += 4) {        // intra: maskA @ V
        v2f a, bb;
        a.x  = Alds[(r0 + lm) * ASTR + j0 + 2 * hi];
        a.y  = Alds[(r0 + lm) * ASTR + j0 + 2 * hi + 1];
        bb.x = Vlds[(j0 + 2 * hi)     * PSTR + c0 + lm];
        bb.y = Vlds[(j0 + 2 * hi + 1) * PSTR + c0 + lm];
        oacc = wmma4(a, bb, oacc);
      }
    }
    __syncthreads();

    // ---- stage G: write output (non-temporal); then update S and s
    {
      int it = wave >> 2, dt = wave & 3;
      for (int e = 0; e < 8; ++e) {
        int row = it * 16 + e + 8 * hi;
        float den = denLds[row];
        den = (den <= 0.f) ? 1.f : den;
        float* dst =
            out + (((size_t)b * Ld + lbase + row) * Hd + h) * Dd + dt * 16 + lm;
        __builtin_nontemporal_store(oacc[e] / den, dst);
      }
    }
    {
      float cs = 0.f;
      for (int r = 0; r < CK; ++r) cs += Kf[r * FSTR + tid];
      sLds[tid] += cs;
    }
    for (int t = wave * 8; t < wave * 8 + 8; ++t) {   // S += Kf^T @ V
      int mt = t >> 2, dt = t & 3;
      int r0 = mt * 16, c0 = dt * 16;
      v8f acc;
      for (int e = 0; e < 8; ++e)
        acc[e] = Slds[(r0 + e + 8 * hi) * PSTR + c0 + lm];
      for (int j0 = 0; j0 < CK; j0 += 4) {
        v2f a, bb;
        a.x  = Kf  [(j0 + 2 * hi)     * FSTR + r0 + lm];
        a.y  = Kf  [(j0 + 2 * hi + 1) * FSTR + r0 + lm];
        bb.x = Vlds[(j0 + 2 * hi)     * PSTR + c0 + lm];
        bb.y = Vlds[(j0 + 2 * hi + 1) * PSTR + c0 + lm];
        acc = wmma4(a, bb, acc);
      }
      for (int e = 0; e < 8; ++e)
        Slds[(r0 + e + 8 * hi) * PSTR + c0 + lm] = acc[e];
    }
    __syncthreads();
  }
}

// ---------------------------------------------------------------------------
extern "C" void kernel_launch(void* const* d_in, const int* in_sizes, int n_in,
                              void* d_out, int out_size, void* d_ws, size_t ws_size,
                              hipStream_t stream) {
  const float* q    = (const float*)d_in[0];   // [B,L,H,D]
  const float* k    = (const float*)d_in[1];   // [B,L,H,D]
  const float* v    = (const float*)d_in[2];   // [B,L,H,D]
  const float* proj = (const float*)d_in[3];   // [M,D]
  float* out = (float*)d_out;                  // [B,L,H,D]
  float* ws  = (float*)d_ws;                   // needs B*H*NSEG*STATE*4 ≈ 17 MB

  (void)in_sizes; (void)n_in; (void)out_size; (void)ws_size;

  perf_attn_segsum<<<Bd * Hd * NSEG, 256, 0, stream>>>(k, v, proj, ws);
  perf_attn_prefix<<<Bd * Hd, 256, 0, stream>>>(ws);
  perf_attn_main  <<<Bd * Hd * NSEG, 256, 0, stream>>>(q, k, v, proj, ws, out);
}